// SAHGPooling_46918222741625
// MI455X (gfx1250) — compile-verified
//
#include <hip/hip_runtime.h>
#include <math.h>

// ---- problem constants (match reference) ----
#define B_   512
#define N_   246
#define D_   256
#define DEG_ 32
#define K_   197          // ceil(0.8*246)
#define NT_  (B_ * N_)    // 125952
#define E_   ((size_t)NT_ * DEG_)  // 4030464
#define BK_  ((size_t)B_ * K_)     // 100864
#define SLOPE_ 0.2f

typedef float v2f __attribute__((ext_vector_type(2)));
typedef float v4f __attribute__((ext_vector_type(4)));
typedef float v8f __attribute__((ext_vector_type(8)));

// ---- flat output layout (floats), reference tuple order ----
static constexpr size_t OFF_XB    = 0;
static constexpr size_t OFF_XD    = OFF_XB + BK_ * D_;
static constexpr size_t OFF_EDGE  = OFF_XD + BK_ * D_;          // 4 blocks of [2E edges][E valid]
static constexpr size_t OFF_BATB  = OFF_EDGE + 12 * E_;
static constexpr size_t OFF_BATD  = OFF_BATB + BK_;
static constexpr size_t OFF_PERMB = OFF_BATD + BK_;
static constexpr size_t OFF_PERMD = OFF_PERMB + BK_;
static constexpr size_t OFF_SB    = OFF_PERMD + BK_;
static constexpr size_t OFF_SD    = OFF_SB + NT_;

// ordered-uint encoding so atomicMax(unsigned) == float max
__device__ __forceinline__ unsigned ordEnc(float f) {
  unsigned u = __float_as_uint(f);
  return (u & 0x80000000u) ? ~u : (u | 0x80000000u);
}
__device__ __forceinline__ float ordDec(unsigned u) {
  unsigned b = (u & 0x80000000u) ? (u ^ 0x80000000u) : ~u;
  return __uint_as_float(b);
}

// ---- 0: scratch init (m=0 encodes minimum under ordEnc; s=num=0; sred=0; masks=-1) ----
__global__ void k_init(float* ws, int* mask) {
  size_t i = (size_t)blockIdx.x * blockDim.x + threadIdx.x;
  ws[2ull * NT_ + i] = 0.0f;              // m[4NT] (as uint 0), s[4NT], num[4NT]
  if (i < 2ull * NT_) mask[i] = -1;
  if (i < 4) ws[14ull * NT_ + i] = 0.0f;  // sred
}

// ---- 1: h = x @ w + b via V_WMMA_F32_16X16X4_F32 ----
// 16 rows per wave, K streamed in chunks of 4. A = 16x4 f32 tile (2 VGPRs/lane),
// B = w chunk broadcast into column 0 only (scalar s_load + branchless cndmask,
// no EXEC divergence inside the WMMA loop). D column 0 lives in lanes 0/16.
__global__ void k_proj_wmma(const float* __restrict__ xb, const float* __restrict__ xd,
                            const float* __restrict__ wb, const float* __restrict__ bb,
                            const float* __restrict__ wd, const float* __restrict__ bd,
                            float* __restrict__ h) {
  const int type = blockIdx.y;
  const float* X = type ? xd : xb;
  const float* W = type ? wd : wb;
  const float bias = type ? bd[0] : bb[0];
  float* H = h + (size_t)type * NT_;
  const int lane = threadIdx.x & 31;
  const int wave = threadIdx.x >> 5;
  const int m  = lane & 15;   // row within 16-row tile
  const int kh = lane >> 4;   // which K-pair {0,1} vs {2,3}
  const bool head = (m == 0); // lanes 0 / 16 carry B column 0
  const size_t rowBase = ((size_t)blockIdx.x * 8 + wave) * 16;
  const float* xrow = X + (rowBase + m) * D_ + 2 * kh;
  const v4f* wq = (const v4f*)W;  // wave-uniform -> scalar loads
  v8f c = {0.f, 0.f, 0.f, 0.f, 0.f, 0.f, 0.f, 0.f};
#pragma unroll 4
  for (int k0 = 0; k0 < D_; k0 += 4) {
    __builtin_prefetch(xrow + k0 + 128, 0, 1);  // global_prefetch_b8, ~512B ahead
    v2f a;
    a.x = xrow[k0];
    a.y = xrow[k0 + 1];
    v4f wv = wq[k0 >> 2];                 // uniform: w[k0..k0+3]
    float blo = kh ? wv.z : wv.x;
    float bhi = kh ? wv.w : wv.y;
    v2f bfrag;
    bfrag.x = head ? blo : 0.f;           // branchless lane select
    bfrag.y = head ? bhi : 0.f;
    c = __builtin_amdgcn_wmma_f32_16x16x4_f32(false, a, false, bfrag, (short)0, c,
                                              false, false);
  }
  if (head) {  // column 0 of D: lane0 holds M=0..7, lane16 holds M=8..15
    union { v8f v; float f[8]; } u; u.v = c;
#pragma unroll
    for (int r = 0; r < 8; ++r) H[rowBase + kh * 8 + r] = u.f[r] + bias;
  }
}

// ---- 2: per-edge logit, segment max over destination (atomic ordered max) ----
__global__ void k_edge_max(const int* ei0, const int* ei1, const int* ei2, const int* ei3,
                           const float* __restrict__ h, const float* __restrict__ att,
                           unsigned* __restrict__ mMax) {
  const int t = blockIdx.y;
  const int* ei = (t == 0) ? ei0 : (t == 1) ? ei1 : (t == 2) ? ei2 : ei3;
  const float* hs = h + ((t < 2) ? 0 : (size_t)NT_);        // src: b,b,d,d
  const float* hd = h + (((t & 1) == 0) ? 0 : (size_t)NT_); // dst: b,d,b,d
  const float aS = att[2 * t], aD = att[2 * t + 1];
  unsigned* M = mMax + (size_t)t * NT_;
  size_t e = (size_t)blockIdx.x * blockDim.x + threadIdx.x;
  if (e >= E_) return;
  int row = ei[e], col = ei[E_ + e];
  float v = hs[row] * aS + hd[col] * aD;
  float l = (v >= 0.f) ? v : SLOPE_ * v;
  atomicMax(&M[col], ordEnc(l));
}

// ---- 3: fused exp / denominator / weighted-numerator accumulation ----
__global__ void k_edge_sum(const int* ei0, const int* ei1, const int* ei2, const int* ei3,
                           const float* __restrict__ h, const float* __restrict__ att,
                           const unsigned* __restrict__ mMax,
                           float* __restrict__ sDen, float* __restrict__ num) {
  const int t = blockIdx.y;
  const int* ei = (t == 0) ? ei0 : (t == 1) ? ei1 : (t == 2) ? ei2 : ei3;
  const float* hs = h + ((t < 2) ? 0 : (size_t)NT_);
  const float* hd = h + (((t & 1) == 0) ? 0 : (size_t)NT_);
  const float aS = att[2 * t], aD = att[2 * t + 1];
  const unsigned* M = mMax + (size_t)t * NT_;
  float* S = sDen + (size_t)t * NT_;
  float* Nu = num + (size_t)t * NT_;
  size_t e = (size_t)blockIdx.x * blockDim.x + threadIdx.x;
  if (e >= E_) return;
  int row = ei[e], col = ei[E_ + e];
  float hsr = hs[row];
  float v = hsr * aS + hd[col] * aD;
  float l = (v >= 0.f) ? v : SLOPE_ * v;
  float ex = expf(l - ordDec(M[col]));
  atomicAdd(&S[col], ex);
  atomicAdd(&Nu[col], hsr * ex);
}

// ---- 4: o = relu(num/(den+eps)); accumulate sum(tanh(o*kw+kb)) per metapath ----
__global__ void k_node_o(float* __restrict__ sDen, float* __restrict__ num,
                         const float* __restrict__ kw, const float* __restrict__ kb,
                         float* __restrict__ sred) {
  size_t i = (size_t)blockIdx.x * blockDim.x + threadIdx.x;  // exact grid == NT
  float kwv = kw[0], kbv = kb[0];
  float tv[4];
#pragma unroll
  for (int t = 0; t < 4; ++t) {
    float o = num[(size_t)t * NT_ + i] / (sDen[(size_t)t * NT_ + i] + 1e-16f);
    o = fmaxf(o, 0.f);
    num[(size_t)t * NT_ + i] = o;  // reuse num as o storage
    tv[t] = tanhf(o * kwv + kbv);
  }
#pragma unroll
  for (int t = 0; t < 4; ++t) {
    float v = tv[t];
    for (int off = 16; off > 0; off >>= 1) v += __shfl_down(v, off, 32);
    if ((threadIdx.x & 31) == 0) atomicAdd(&sred[t], v);
  }
}

// ---- 5: semantic-attention weights + final node scores ----
__global__ void k_score(const float* __restrict__ o, const float* __restrict__ sred,
                        const float* __restrict__ q, float* __restrict__ out) {
  size_t i = (size_t)blockIdx.x * blockDim.x + threadIdx.x;  // exact grid == NT
  float qv = q[0];
  const float invNT = 1.0f / (float)NT_;
  // bold group: stack(o_bb, o_db) -> types 0, 2
  float e0 = qv * (sred[0] * invNT), e1 = qv * (sred[2] * invNT);
  float mx = fmaxf(e0, e1);
  float w0 = expf(e0 - mx), w1 = expf(e1 - mx);
  float inv = 1.f / (w0 + w1);
  out[OFF_SB + i] = (w0 * o[i] + w1 * o[2ull * NT_ + i]) * inv;
  // dti group: stack(o_bd, o_dd) -> types 1, 3
  float f0 = qv * (sred[1] * invNT), f1 = qv * (sred[3] * invNT);
  float mx2 = fmaxf(f0, f1);
  float u0 = expf(f0 - mx2), u1 = expf(f1 - mx2);
  float inv2 = 1.f / (u0 + u1);
  out[OFF_SD + i] = (u0 * o[1ull * NT_ + i] + u1 * o[3ull * NT_ + i]) * inv2;
}

// ---- 6: per-graph top-K via rank counting (descending, ties -> lower index) ----
__global__ void k_topk(const float* __restrict__ outR,
                       const int* __restrict__ batch_b, const int* __restrict__ batch_d,
                       int* __restrict__ permI, float* __restrict__ out) {
  const int type = blockIdx.y;
  const int b = blockIdx.x;
  const float* s = outR + (type ? OFF_SD : OFF_SB) + (size_t)b * N_;
  __shared__ float sc[N_];
  int i = threadIdx.x;
  if (i < N_) sc[i] = s[i];
  __syncthreads();
  if (i < N_) {
    float si = sc[i];
    int rank = 0;
    for (int j = 0; j < N_; ++j) {
      float sj = sc[j];
      rank += (sj > si) || (sj == si && j < i);
    }
    if (rank < K_) {
      int node = b * N_ + i;
      size_t g = (size_t)b * K_ + rank;
      permI[(size_t)type * BK_ + g] = node;
      out[(type ? OFF_PERMD : OFF_PERMB) + g] = (float)node;
      const int* bat = type ? batch_d : batch_b;
      out[(type ? OFF_BATD : OFF_BATB) + g] = (float)bat[node];
    }
  }
}

// ---- 7: mask[perm[i]] = i ----
__global__ void k_mask(const int* __restrict__ permI, int* __restrict__ mask) {
  const int type = blockIdx.y;
  size_t i = (size_t)blockIdx.x * blockDim.x + threadIdx.x;  // exact grid == BK
  int p = permI[(size_t)type * BK_ + i];
  mask[(size_t)type * NT_ + p] = (int)i;
}

// ---- 8: xb/xd = x[perm] * tanh(s[perm]) ----
__global__ void k_gather(const float* __restrict__ xb, const float* __restrict__ xd,
                         const int* __restrict__ permI, float* __restrict__ out) {
  const int type = blockIdx.y;
  size_t j = blockIdx.x;
  int p = permI[(size_t)type * BK_ + j];
  float g = tanhf(out[(type ? OFF_SD : OFF_SB) + p]);
  const float* X = type ? xd : xb;
  out[(type ? OFF_XD : OFF_XB) + j * D_ + threadIdx.x] =
      X[(size_t)p * D_ + threadIdx.x] * g;
}

// ---- 9: filter_adj: remap edges through masks, -1 / validity flags ----
__global__ void k_filter(const int* ei0, const int* ei1, const int* ei2, const int* ei3,
                         const int* __restrict__ mask, float* __restrict__ out) {
  const int t = blockIdx.y;
  const int* ei = (t == 0) ? ei0 : (t == 1) ? ei1 : (t == 2) ? ei2 : ei3;
  const int* mr = mask + ((t < 2) ? 0 : (size_t)NT_);        // row node type
  const int* mc = mask + (((t & 1) == 0) ? 0 : (size_t)NT_); // col node type
  size_t e = (size_t)blockIdx.x * blockDim.x + threadIdx.x;
  if (e >= E_) return;
  int r = mr[ei[e]];
  int c = mc[ei[E_ + e]];
  bool valid = (r >= 0) && (c >= 0);
  size_t base = OFF_EDGE + (size_t)t * 3 * E_;
  out[base + e]            = valid ? (float)r : -1.f;
  out[base + E_ + e]       = valid ? (float)c : -1.f;
  out[base + 2 * E_ + e]   = valid ? 1.f : 0.f;
}

extern "C" void kernel_launch(void* const* d_in, const int* in_sizes, int n_in,
                              void* d_out, int out_size, void* d_ws, size_t ws_size,
                              hipStream_t stream) {
  const float* x_bold = (const float*)d_in[0];
  const float* x_dti  = (const float*)d_in[1];
  const float* pw_b   = (const float*)d_in[2];
  const float* pb_b   = (const float*)d_in[3];
  const float* pw_d   = (const float*)d_in[4];
  const float* pb_d   = (const float*)d_in[5];
  const float* att    = (const float*)d_in[6];
  const float* klw    = (const float*)d_in[7];
  const float* klb    = (const float*)d_in[8];
  const float* q      = (const float*)d_in[9];
  const int* ei0 = (const int*)d_in[10];  // bb
  const int* ei1 = (const int*)d_in[11];  // bd
  const int* ei2 = (const int*)d_in[12];  // db
  const int* ei3 = (const int*)d_in[13];  // dd
  const int* batch_b = (const int*)d_in[14];
  const int* batch_d = (const int*)d_in[15];
  float* out = (float*)d_out;

  // workspace layout (floats): h[2NT] | m[4NT] | s[4NT] | num/o[4NT] | sred[4] | permI[2BK] | mask[2NT]
  float* ws      = (float*)d_ws;
  float* h       = ws;
  unsigned* mMax = (unsigned*)(ws + 2ull * NT_);
  float* sDen    = ws + 6ull * NT_;
  float* num     = ws + 10ull * NT_;
  float* sred    = ws + 14ull * NT_;
  int* permI     = (int*)(ws + 14ull * NT_ + 4);
  int* mask      = permI + 2ull * BK_;

  const int edgeBlocks = (int)(E_ / 256);  // 15744 exact

  k_init<<<dim3((unsigned)(12ull * NT_ / 256)), 256, 0, stream>>>(ws, mask);
  k_proj_wmma<<<dim3(NT_ / 128, 2), 256, 0, stream>>>(x_bold, x_dti, pw_b, pb_b,
                                                      pw_d, pb_d, h);
  k_edge_max<<<dim3(edgeBlocks, 4), 256, 0, stream>>>(ei0, ei1, ei2, ei3, h, att, mMax);
  k_edge_sum<<<dim3(edgeBlocks, 4), 256, 0, stream>>>(ei0, ei1, ei2, ei3, h, att, mMax,
                                                      sDen, num);
  k_node_o<<<dim3(NT_ / 256), 256, 0, stream>>>(sDen, num, klw, klb, sred);
  k_score<<<dim3(NT_ / 256), 256, 0, stream>>>(num, sred, q, out);
  k_topk<<<dim3(B_, 2), 256, 0, stream>>>(out, batch_b, batch_d, permI, out);
  k_mask<<<dim3((unsigned)(BK_ / 256), 2), 256, 0, stream>>>(permI, mask);
  k_gather<<<dim3((unsigned)BK_, 2), 256, 0, stream>>>(x_bold, x_dti, permI, out);
  k_filter<<<dim3(edgeBlocks, 4), 256, 0, stream>>>(ei0, ei1, ei2, ei3, mask, out);
}